// MDTA_86526411145373
// MI455X (gfx1250) — compile-verified
//
#include <hip/hip_runtime.h>
#include <math.h>

typedef __attribute__((ext_vector_type(2))) float v2f;
typedef __attribute__((ext_vector_type(8))) float v8f;

#define NB    8
#define CIN   48
#define C3    144
#define HEADS 8
#define DH    6
#define IMG   256
#define HW    65536
#define EPSN  1e-12f
#define ATT_SCALE 0.40824829046386296f   /* 6^-0.5 */

// CDNA5 direct global->LDS async copy path (guarded: falls back to
// load/ds_store staging if this toolchain doesn't declare the builtins).
#if defined(__has_builtin)
#  if __has_builtin(__builtin_amdgcn_global_load_async_to_lds_b32) && \
      __has_builtin(__builtin_amdgcn_s_wait_asynccnt)
#    define USE_ASYNC_LDS 1
#  endif
#endif
#ifndef USE_ASYNC_LDS
#  define USE_ASYNC_LDS 0
#endif

#if USE_ASYNC_LDS
typedef __attribute__((address_space(1))) int gas_int;   // global (__device__) int
typedef __attribute__((address_space(3))) int las_int;   // LDS (__shared__) int
#endif

// ---------------------------------------------------------------------------
// K1: 1x1 pointwise conv as GEMM  qkv[b,144,n] = W(144x48) @ x[b,48,n]
// One wave = 16 pixels, 9 M-tiles of 16 output channels, K=48 in steps of 4.
// A(16x4 f32) layout: lanes 0-15 hold M rows with K={0,1} in the two VGPRs,
// lanes 16-31 hold K={2,3}.  B(4x16): lanes 0-15 = N cols of K rows {0,1},
// lanes 16-31 = K rows {2,3}.  C/D(16x16): VGPR g -> row g (+8 for high lanes).
// ---------------------------------------------------------------------------
__global__ __launch_bounds__(256) void k_qkv_pw(const float* __restrict__ x,
                                                const float* __restrict__ w_qkv,
                                                float* __restrict__ qkv) {
    __shared__ float sW[C3 * CIN];                      // 27 KB
    for (int i = threadIdx.x; i < C3 * CIN; i += 256) sW[i] = w_qkv[i];
    __syncthreads();

    const int wave  = threadIdx.x >> 5;
    const int lane  = threadIdx.x & 31;
    const int col   = lane & 15;        // pixel within 16-wide tile / M row id
    const int khalf = lane >> 4;        // selects K pair {0,1} vs {2,3}

    const int idx = blockIdx.x * 8 + wave;   // 16-pixel group id (4096/batch)
    const int b   = idx >> 12;
    const int n0  = (idx & 4095) << 4;

    const float* xb = x + ((size_t)b * CIN) * HW + n0 + col;

    v8f acc[9];
#pragma unroll
    for (int m = 0; m < 9; ++m) acc[m] = (v8f)0.0f;

#pragma unroll
    for (int ks = 0; ks < 12; ++ks) {
        const int k0 = ks * 4 + khalf * 2;
        v2f bf;
        bf.x = xb[(size_t)(k0 + 0) * HW];
        bf.y = xb[(size_t)(k0 + 1) * HW];
#pragma unroll
        for (int m = 0; m < 9; ++m) {
            v2f af;
            af.x = sW[(m * 16 + col) * CIN + k0 + 0];
            af.y = sW[(m * 16 + col) * CIN + k0 + 1];
            acc[m] = __builtin_amdgcn_wmma_f32_16x16x4_f32(
                false, af, false, bf, (short)0, acc[m], false, false);
        }
    }

    float* ob = qkv + ((size_t)b * C3) * HW + n0 + col;
#pragma unroll
    for (int m = 0; m < 9; ++m)
#pragma unroll
        for (int g = 0; g < 8; ++g)
            ob[(size_t)(m * 16 + g + khalf * 8) * HW] = acc[m][g];
}

// ---------------------------------------------------------------------------
// K2: 3x3 depthwise conv ('SAME', zero pad).  One block = one channel-image,
// 64x4 pixel tile staged in LDS with 1-pixel halo.  Pure bandwidth.
// Uses the CDNA5 async global->LDS DMA path when available (ASYNCcnt).
// ---------------------------------------------------------------------------
#define TX 64
#define TY 4
__global__ __launch_bounds__(256) void k_dw(const float* __restrict__ qkv,
                                            const float* __restrict__ w_dw,
                                            float* __restrict__ qkv_dw) {
    __shared__ float s[(TY + 2) * (TX + 2)];
    const int tile = blockIdx.x & 255;         // 4 x-tiles * 64 y-tiles
    const int bc   = blockIdx.x >> 8;          // 0 .. NB*C3-1
    const int ch   = bc % C3;
    const int x0   = (tile & 3) * TX;
    const int y0   = (tile >> 2) * TY;

    const float* in = qkv + (size_t)bc * HW;

#if USE_ASYNC_LDS
    for (int i = threadIdx.x; i < (TY + 2) * (TX + 2); i += 256) {
        const int ly = i / (TX + 2), lx = i % (TX + 2);
        const int gy = y0 + ly - 1, gx = x0 + lx - 1;
        if (gy >= 0 && gy < IMG && gx >= 0 && gx < IMG) {
            __builtin_amdgcn_global_load_async_to_lds_b32(
                (gas_int*)(in + gy * IMG + gx),
                (las_int*)(&s[i]),
                0, 0);
        } else {
            s[i] = 0.0f;                // zero padding (DScnt-ordered)
        }
    }
    __builtin_amdgcn_s_wait_asynccnt(0);
    __syncthreads();
#else
    for (int i = threadIdx.x; i < (TY + 2) * (TX + 2); i += 256) {
        const int ly = i / (TX + 2), lx = i % (TX + 2);
        const int gy = y0 + ly - 1, gx = x0 + lx - 1;
        float v = 0.0f;
        if (gy >= 0 && gy < IMG && gx >= 0 && gx < IMG) v = in[gy * IMG + gx];
        s[i] = v;
    }
    __syncthreads();
#endif

    float w[9];
#pragma unroll
    for (int i = 0; i < 9; ++i) w[i] = w_dw[ch * 9 + i];

    const int tx = threadIdx.x & (TX - 1);
    const int ty = threadIdx.x / TX;
    float acc = 0.0f;
#pragma unroll
    for (int ky = 0; ky < 3; ++ky)
#pragma unroll
        for (int kx = 0; kx < 3; ++kx)
            acc += w[ky * 3 + kx] * s[(ty + ky) * (TX + 2) + tx + kx];

    qkv_dw[(size_t)bc * HW + (y0 + ty) * IMG + x0 + tx] = acc;
}

// ---------------------------------------------------------------------------
// K3a: per (b,head) reduce over n=65536: Gram q.k^T (6x6), sum(q^2), sum(k^2).
// 48 accumulators/thread, wave32 shfl_xor reduce, LDS atomic combine.
// ---------------------------------------------------------------------------
__global__ __launch_bounds__(256) void k_stats(const float* __restrict__ qkv_dw,
                                               float* __restrict__ stats) {
    const int b = blockIdx.x >> 3;
    const int h = blockIdx.x & 7;
    const float* qb = qkv_dw + ((size_t)b * C3 + h * DH) * HW;
    const float* kb = qkv_dw + ((size_t)b * C3 + CIN + h * DH) * HW;

    float acc[48];
#pragma unroll
    for (int i = 0; i < 48; ++i) acc[i] = 0.0f;

    for (int n = threadIdx.x; n < HW; n += 256) {
        float qv[DH], kv[DH];
#pragma unroll
        for (int d = 0; d < DH; ++d) {
            qv[d] = qb[(size_t)d * HW + n];
            kv[d] = kb[(size_t)d * HW + n];
        }
#pragma unroll
        for (int d = 0; d < DH; ++d) {
#pragma unroll
            for (int e = 0; e < DH; ++e) acc[d * DH + e] += qv[d] * kv[e];
            acc[36 + d] += qv[d] * qv[d];
            acc[42 + d] += kv[d] * kv[d];
        }
    }
#pragma unroll
    for (int i = 0; i < 48; ++i)
        for (int off = 16; off; off >>= 1)
            acc[i] += __shfl_xor(acc[i], off, 32);

    __shared__ float sred[48];
    if (threadIdx.x < 48) sred[threadIdx.x] = 0.0f;
    __syncthreads();
    if ((threadIdx.x & 31) == 0)
        for (int i = 0; i < 48; ++i) atomicAdd(&sred[i], acc[i]);
    __syncthreads();
    if (threadIdx.x < 48)
        stats[blockIdx.x * 48 + threadIdx.x] = sred[threadIdx.x];
}

// ---------------------------------------------------------------------------
// K3b: attn = softmax(Gram/(||q||*||k||)*scale); M_b = W_proj @ blockdiag(attn)
// ---------------------------------------------------------------------------
__global__ __launch_bounds__(256) void k_softmax_proj(const float* __restrict__ stats,
                                                      const float* __restrict__ w_proj,
                                                      float* __restrict__ Mout) {
    const int b = blockIdx.x;
    __shared__ float sstat[HEADS * 48];
    __shared__ float sattn[HEADS * 36];
    for (int i = threadIdx.x; i < HEADS * 48; i += 256)
        sstat[i] = stats[b * HEADS * 48 + i];
    __syncthreads();

    if (threadIdx.x < 48) {
        const int h = threadIdx.x / DH, d = threadIdx.x % DH;
        const float nq = fmaxf(sqrtf(sstat[h * 48 + 36 + d]), EPSN);
        float row[DH];
        float mx = -1e30f;
#pragma unroll
        for (int e = 0; e < DH; ++e) {
            const float nk = fmaxf(sqrtf(sstat[h * 48 + 42 + e]), EPSN);
            row[e] = sstat[h * 48 + d * DH + e] / (nq * nk) * ATT_SCALE;
            mx = fmaxf(mx, row[e]);
        }
        float sum = 0.0f;
#pragma unroll
        for (int e = 0; e < DH; ++e) { row[e] = __expf(row[e] - mx); sum += row[e]; }
#pragma unroll
        for (int e = 0; e < DH; ++e) sattn[h * 36 + d * DH + e] = row[e] / sum;
    }
    __syncthreads();

    for (int i = threadIdx.x; i < 48 * 48; i += 256) {
        const int co = i / 48, cc = i % 48;
        const int h = cc / DH, e = cc % DH;
        float sum = 0.0f;
#pragma unroll
        for (int d = 0; d < DH; ++d)
            sum += w_proj[co * 48 + h * DH + d] * sattn[h * 36 + d * DH + e];
        Mout[b * 48 * 48 + i] = sum;
    }
}

// ---------------------------------------------------------------------------
// K4: out[b] = M_b(48x48) @ v_dw[b](48x65536) via WMMA; writes final output.
// ---------------------------------------------------------------------------
__global__ __launch_bounds__(256) void k_out_gemm(const float* __restrict__ qkv_dw,
                                                  const float* __restrict__ M,
                                                  float* __restrict__ out) {
    __shared__ float sM[48 * 48];
    const int b = blockIdx.x >> 9;                   // 512 blocks per batch
    for (int i = threadIdx.x; i < 48 * 48; i += 256) sM[i] = M[b * 48 * 48 + i];
    __syncthreads();

    const int wave  = threadIdx.x >> 5;
    const int lane  = threadIdx.x & 31;
    const int col   = lane & 15;
    const int khalf = lane >> 4;
    const int n0    = (((blockIdx.x & 511) * 8) + wave) * 16;

    const float* vb = qkv_dw + ((size_t)b * C3 + 2 * CIN) * HW + n0 + col;

    v8f acc[3];
#pragma unroll
    for (int m = 0; m < 3; ++m) acc[m] = (v8f)0.0f;

#pragma unroll
    for (int ks = 0; ks < 12; ++ks) {
        const int k0 = ks * 4 + khalf * 2;
        v2f bf;
        bf.x = vb[(size_t)(k0 + 0) * HW];
        bf.y = vb[(size_t)(k0 + 1) * HW];
#pragma unroll
        for (int m = 0; m < 3; ++m) {
            v2f af;
            af.x = sM[(m * 16 + col) * 48 + k0 + 0];
            af.y = sM[(m * 16 + col) * 48 + k0 + 1];
            acc[m] = __builtin_amdgcn_wmma_f32_16x16x4_f32(
                false, af, false, bf, (short)0, acc[m], false, false);
        }
    }

    float* ob = out + ((size_t)b * CIN) * HW + n0 + col;
#pragma unroll
    for (int m = 0; m < 3; ++m)
#pragma unroll
        for (int g = 0; g < 8; ++g)
            ob[(size_t)(m * 16 + g + khalf * 8) * HW] = acc[m][g];
}

// ---------------------------------------------------------------------------
extern "C" void kernel_launch(void* const* d_in, const int* in_sizes, int n_in,
                              void* d_out, int out_size, void* d_ws, size_t ws_size,
                              hipStream_t stream) {
    const float* x      = (const float*)d_in[0];
    const float* w_qkv  = (const float*)d_in[1];
    const float* w_dw   = (const float*)d_in[2];
    const float* w_proj = (const float*)d_in[3];
    float* out = (float*)d_out;

    float* ws     = (float*)d_ws;
    float* qkv    = ws;                                  // NB*C3*HW floats
    float* qkv_dw = qkv + (size_t)NB * C3 * HW;          // NB*C3*HW floats
    float* stats  = qkv_dw + (size_t)NB * C3 * HW;       // 64*48 floats
    float* Mmat   = stats + 64 * 48;                     // NB*48*48 floats

    k_qkv_pw<<<4096, 256, 0, stream>>>(x, w_qkv, qkv);
    k_dw<<<NB * C3 * 256, 256, 0, stream>>>(qkv, w_dw, qkv_dw);
    k_stats<<<64, 256, 0, stream>>>(qkv_dw, stats);
    k_softmax_proj<<<NB, 256, 0, stream>>>(stats, w_proj, Mmat);
    k_out_gemm<<<4096, 256, 0, stream>>>(qkv_dw, Mmat, out);
}